// ConvOffset3d_52012053954779
// MI455X (gfx1250) — compile-verified
//
#include <hip/hip_runtime.h>

typedef float v2f __attribute__((ext_vector_type(2)));
typedef float v4f __attribute__((ext_vector_type(4)));
typedef float v8f __attribute__((ext_vector_type(8)));

#define BB        2
#define CIN       64
#define COUT      64
#define DD        8
#define HH        32
#define WW        32
#define KK        27
#define PP        8192      // DOUT*HOUT*WOUT = 8*32*32
#define CH_STRIDE 8192      // D*H*W elements per (b,c) slice
#define TP        64        // output positions per block
#define SPITCH    68        // padded LDS row pitch for S[p][c] (CIN + 4)

// ---------------------------------------------------------------------------
// Prologue: reorder weight [COUT][CIN][K] -> Wt[K][COUT][CIN] so GEMM
// A-fragments are contiguous 8B (v2f) loads per lane.
// ---------------------------------------------------------------------------
__global__ void reorder_weight_kernel(const float* __restrict__ w,
                                      float* __restrict__ wt) {
  int t = blockIdx.x * blockDim.x + threadIdx.x;
  if (t >= KK * COUT * CIN) return;
  int c = t % CIN;
  int o = (t / CIN) % COUT;
  int k = t / (CIN * COUT);
  wt[(k * COUT + o) * CIN + c] = w[(o * CIN + c) * KK + k];
}

// ---------------------------------------------------------------------------
// Fused deformable-conv3d:
//   per block: batch b, 64 output positions; loops 27 taps:
//     corner calc (64 lanes) -> channel gather S[64p][64c] (transposed, so
//     B-fragments are single ds_load_b64) -> f32 WMMA 16x16x4, two column
//     tiles per wave sharing every A-fragment load.
// ---------------------------------------------------------------------------
__global__ __launch_bounds__(256)
void deform_conv3d_wmma_kernel(const float* __restrict__ x,
                               const float* __restrict__ offset,
                               const float* __restrict__ wt,
                               const float* __restrict__ bias,
                               float* __restrict__ out) {
  __shared__ float s_S[TP * SPITCH];    // sampled tile S[p][c], padded pitch
  __shared__ float s_cw[TP * 8];        // 8 trilinear corner weights per p
  __shared__ int   s_cidx[TP * 8];      // 8 clamped flat corner indices per p

  const int tid  = threadIdx.x;
  const int lane = tid & 31;
  const int wave = tid >> 5;
  const int half = lane >> 4;           // 0: lanes 0-15, 1: lanes 16-31
  const int m    = lane & 15;

  const int b     = blockIdx.x >> 7;    // 128 p-tiles per batch
  const int pbase = (blockIdx.x & 127) << 6;

  const float* xb = x + b * (CIN * CH_STRIDE);

  // wave -> one 16-row COUT strip x two 16-col p sub-tiles
  const int o0      = (wave & 3) << 4;  // COUT strip
  const int colbase = (wave >> 2) << 5; // 0 or 32 within the 64-wide p tile

  // gather assignment: thread handles 16 consecutive channels for one p
  const int p_local = tid & 63;
  const int cbase   = (tid >> 6) << 4;

  v8f acc0 = {};                        // f32 C/D accumulators (8 VGPRs each)
  v8f acc1 = {};

  for (int k = 0; k < KK; ++k) {
    __syncthreads();                    // protect LDS from previous tap

    // ---- corner computation: one lane per output position --------------
    if (tid < TP) {
      int p  = pbase + tid;
      int od = p >> 10, oh = (p >> 5) & 31, ow = p & 31;
      int kd = k / 9, kh = (k / 3) % 3, kw = k % 3;
      // offset layout: [b][k][axis][p], axis stride = PP
      int obase = ((b * KK + k) * 3) * PP + p;
      float zd = (float)(od - 1 + kd) + offset[obase];
      float zh = (float)(oh - 1 + kh) + offset[obase + PP];
      float zw = (float)(ow - 1 + kw) + offset[obase + 2 * PP];
      float d0 = floorf(zd), h0 = floorf(zh), w0 = floorf(zw);
      float fd = zd - d0, fh = zh - h0, fw = zw - w0;
      int d0i = (int)d0, h0i = (int)h0, w0i = (int)w0;
#pragma unroll
      for (int j = 0; j < 8; ++j) {
        int dd = (j >> 2) & 1, dh = (j >> 1) & 1, dw = j & 1;
        int di = d0i + dd, hi = h0i + dh, wi = w0i + dw;
        bool valid = (di >= 0) && (di < DD) && (hi >= 0) && (hi < HH) &&
                     (wi >= 0) && (wi < WW);
        float wgt = (dd ? fd : 1.0f - fd) * (dh ? fh : 1.0f - fh) *
                    (dw ? fw : 1.0f - fw);
        int ci = min(max(di, 0), DD - 1) * (HH * WW) +
                 min(max(hi, 0), HH - 1) * WW +
                 min(max(wi, 0), WW - 1);
        s_cw[tid * 8 + j]   = valid ? wgt : 0.0f;
        s_cidx[tid * 8 + j] = ci;
      }
    }
    __syncthreads();

    // ---- channel gather: amortize corner math over all 64 channels -----
    {
      float w8[8];
      int   i8[8];
#pragma unroll
      for (int j = 0; j < 8; ++j) {
        w8[j] = s_cw[p_local * 8 + j];
        i8[j] = s_cidx[p_local * 8 + j];
      }
#pragma unroll
      for (int q = 0; q < 4; ++q) {     // 4 x float4 = 16 channels
        v4f v4;
#pragma unroll
        for (int cc = 0; cc < 4; ++cc) {
          const float* xc = xb + (cbase + q * 4 + cc) * CH_STRIDE;
          float v = 0.0f;
#pragma unroll
          for (int j = 0; j < 8; ++j) v += w8[j] * xc[i8[j]];
          v4[cc] = v;
        }
        *(v4f*)&s_S[p_local * SPITCH + cbase + q * 4] = v4;
      }
    }
    __syncthreads();

    // ---- WMMA phase: 16 K-steps of 4 over CIN, 2 col tiles share A -----
    const float* wk = wt + k * (COUT * CIN);
#pragma unroll
    for (int cs = 0; cs < 16; ++cs) {
      int c = cs * 4 + 2 * half;  // lanes 0-15: K=c,c+1 ; lanes 16-31: K=c+2,c+3
      // A fragment (16x4 f32): row o0+m, two consecutive channels -> b64 load
      v2f a = *(const v2f*)(wk + (o0 + m) * CIN + c);
      // B fragments (4x16 f32): transposed LDS => contiguous ds_load_b64
      v2f b0 = *(const v2f*)&s_S[(colbase + m) * SPITCH + c];
      v2f b1 = *(const v2f*)&s_S[(colbase + 16 + m) * SPITCH + c];
      acc0 = __builtin_amdgcn_wmma_f32_16x16x4_f32(
          false, a, false, b0, (short)0, acc0, false, false);
      acc1 = __builtin_amdgcn_wmma_f32_16x16x4_f32(
          false, a, false, b1, (short)0, acc1, false, false);
    }
  }

  // ---- epilogue: bias + coalesced stores (two 16x16 tiles) --------------
  const int p0 = pbase + colbase + m;
#pragma unroll
  for (int i = 0; i < 8; ++i) {
    int o = o0 + i + 8 * half;    // C/D layout: VGPR i -> M = i + 8*half
    float bo = bias[o];
    float* orow = out + (b * COUT + o) * PP;
    orow[p0]      = acc0[i] + bo;
    orow[p0 + 16] = acc1[i] + bo;
  }
}

// ---------------------------------------------------------------------------
extern "C" void kernel_launch(void* const* d_in, const int* in_sizes, int n_in,
                              void* d_out, int out_size, void* d_ws,
                              size_t ws_size, hipStream_t stream) {
  const float* x      = (const float*)d_in[0];
  const float* offset = (const float*)d_in[1];
  const float* weight = (const float*)d_in[2];
  const float* bias   = (const float*)d_in[3];
  float* out = (float*)d_out;
  float* wt  = (float*)d_ws;   // KK*COUT*CIN floats = 432 KB

  int nw = KK * COUT * CIN;
  reorder_weight_kernel<<<(nw + 255) / 256, 256, 0, stream>>>(weight, wt);

  // B * (PP/64) = 2 * 128 = 256 blocks, 256 threads (8 wave32) each
  deform_conv3d_wmma_kernel<<<256, 256, 0, stream>>>(x, offset, wt, bias, out);
}